// P2RRegionLoss_58128087384436
// MI455X (gfx1250) — compile-verified
//
#include <hip/hip_runtime.h>
#include <hip/hip_bf16.h>

// ---------------- problem constants (fixed by the reference) ----------------
constexpr int BATCH = 64;
constexpr int Hdim  = 512;
constexpr int Wdim  = 512;
constexpr int HW    = Hdim * Wdim;          // 262144 pixels / batch
constexpr int NPTS  = 4000;
constexpr float EPSF     = 1e-6f;
constexpr float COUNT_W  = 2.0f;
constexpr float SPATIAL_W= 0.15f;
constexpr float SCALE_W  = 0.5f;

// histogram: 2 uint16 cells packed per uint32 word
constexpr int HIST_WORDS      = BATCH * (HW / 2);        // 8,388,608 words = 32 MiB
constexpr int BLOCKS_PER_BATCH= 32;
constexpr int PIX_PER_BLOCK   = HW / BLOCKS_PER_BATCH;   // 8192
constexpr int RED_THREADS     = 256;
constexpr int RED_ITERS       = PIX_PER_BLOCK / (RED_THREADS * 4); // 8

typedef __attribute__((ext_vector_type(2))) float v2f;
typedef __attribute__((ext_vector_type(8))) float v8f;

// ---------------- WMMA-based exact f32 wave reduction ----------------
// D = ones(16x4) x B(4x16) + 0 : every row of D is the vector of 16 column
// sums; each of the 64 B-slots (32 lanes x 2 VGPRs) feeds exactly one column.
// We put the lane value in B VGPR0 and 0 in VGPR1, so column sums fold
// 32 lanes -> 16 columns in ONE v_wmma_f32_16x16x4_f32; 4 xor-shuffles then
// sum the 16 columns. Exact in f32 (just a re-association of the adds).
__device__ __forceinline__ float wave_reduce_wmma(float v) {
  v2f a = {1.0f, 1.0f};      // A = all-ones 16x4
  v2f b = {v, 0.0f};         // one payload slot per lane, other slot = 0
  v8f c = {};                // C = 0
  v8f d = __builtin_amdgcn_wmma_f32_16x16x4_f32(
      /*neg_a=*/false, a, /*neg_b=*/false, b,
      /*c_mod=*/(short)0, c, /*reuse_a=*/false, /*reuse_b=*/false);
  float s = d[0];            // lane L holds colsum(L & 15) (rows identical)
  s += __shfl_xor(s, 1, 32);
  s += __shfl_xor(s, 2, 32);
  s += __shfl_xor(s, 4, 32);
  s += __shfl_xor(s, 8, 32); // all lanes: sum over all 16 columns = wave sum
  return s;
}

// ---------------- kernel 1: zero the packed histogram ----------------
__global__ void zero_hist_kernel(uint4* __restrict__ hist4) {
  // HIST_WORDS/4 = 2,097,152 uint4 stores; 2048 blocks x 256 thr x 4 iters
  const unsigned stride = gridDim.x * blockDim.x;           // 524288
  unsigned idx = blockIdx.x * blockDim.x + threadIdx.x;
  const uint4 z = {0u, 0u, 0u, 0u};
#pragma unroll
  for (int k = 0; k < 4; ++k) {
    hist4[idx] = z;
    idx += stride;
  }
}

// ---------------- kernel 2: scatter points into packed histogram ------------
__global__ void scatter_points_kernel(const int* __restrict__ points,
                                      const int* __restrict__ downscale,
                                      unsigned* __restrict__ hist) {
  int t = blockIdx.x * blockDim.x + threadIdx.x;            // one point each
  if (t >= BATCH * NPTS) return;
  const int b  = t / NPTS;
  const int ds = downscale[0];
  const int x  = points[2 * t + 0];
  const int y  = points[2 * t + 1];
  int fx = x / ds; fx = fx < 0 ? 0 : (fx > Wdim - 1 ? Wdim - 1 : fx);
  int fy = y / ds; fy = fy < 0 ? 0 : (fy > Hdim - 1 ? Hdim - 1 : fy);
  const int lin = fy * Wdim + fx;                           // cell in [0, HW)
  // two uint16 cells per word; counts <= 4000 -> no carry across halves
  atomicAdd(hist + b * (HW / 2) + (lin >> 1), 1u << (16 * (lin & 1)));
}

// ---------------- kernel 3: fused single-pass reduction ----------------
// Per block: contiguous 8192-pixel slice of one batch.
// Accumulate S1 = sum p, S2 = sum p^2, Spc = sum p*c, Scc = sum c^2.
__global__ void fused_reduce_kernel(const float* __restrict__ pred,
                                    const unsigned* __restrict__ hist,
                                    float* __restrict__ partials) {
  const int batch = blockIdx.x / BLOCKS_PER_BATCH;
  const int blk   = blockIdx.x % BLOCKS_PER_BATCH;
  const int tid   = threadIdx.x;

  const float4* __restrict__ p4 =
      (const float4*)(pred + (size_t)batch * HW + (size_t)blk * PIX_PER_BLOCK);
  const uint2* __restrict__ h2 =
      (const uint2*)(hist + (size_t)batch * (HW / 2) + (size_t)blk * (PIX_PER_BLOCK / 2));

  float s1 = 0.f, s2 = 0.f, spc = 0.f, scc = 0.f;
#pragma unroll
  for (int it = 0; it < RED_ITERS; ++it) {
    const int i = it * RED_THREADS + tid;   // float4 covers pixels 4i..4i+3
    const float4 p = p4[i];
    const uint2  w = h2[i];                 // 2 words = 4 packed uint16 counts
    const float c0 = (float)(w.x & 0xFFFFu);
    const float c1 = (float)(w.x >> 16);
    const float c2 = (float)(w.y & 0xFFFFu);
    const float c3 = (float)(w.y >> 16);
    s1  += (p.x + p.y) + (p.z + p.w);
    s2  += (p.x * p.x + p.y * p.y) + (p.z * p.z + p.w * p.w);
    spc += (p.x * c0 + p.y * c1) + (p.z * c2 + p.w * c3);
    scc += (c0 * c0 + c1 * c1) + (c2 * c2 + c3 * c3);
  }

  // wave-level reduction on the matrix unit (EXEC is all-ones here: no
  // divergence anywhere in this kernel, full 256-thread blocks)
  s1  = wave_reduce_wmma(s1);
  s2  = wave_reduce_wmma(s2);
  spc = wave_reduce_wmma(spc);
  scc = wave_reduce_wmma(scc);

  __shared__ float wpart[RED_THREADS / 32][4];
  const int wave = tid >> 5, lane = tid & 31;
  if (lane == 0) {
    wpart[wave][0] = s1; wpart[wave][1] = s2;
    wpart[wave][2] = spc; wpart[wave][3] = scc;
  }
  __syncthreads();
  if (tid < 4) {                            // fixed-order, deterministic
    float acc = 0.f;
#pragma unroll
    for (int wv = 0; wv < RED_THREADS / 32; ++wv) acc += wpart[wv][tid];
    partials[(size_t)blockIdx.x * 4 + tid] = acc;
  }
}

// ---------------- kernel 4: finalize the four scalars ----------------
__global__ void finalize_kernel(const float* __restrict__ partials,
                                float* __restrict__ out) {
  const int t = threadIdx.x;                // t = batch index, 64 threads
  float S0 = 0.f, S1 = 0.f, S2 = 0.f, S3 = 0.f;
  for (int i = 0; i < BLOCKS_PER_BATCH; ++i) {
    const float* p = partials + ((size_t)t * BLOCKS_PER_BATCH + i) * 4;
    S0 += p[0]; S1 += p[1]; S2 += p[2]; S3 += p[3];
  }
  const float gt  = (float)NPTS;
  const float dg  = gt + EPSF;              // gt.sum == N exactly (all clipped in-range)
  const float pc  = S0 + EPSF;
  const float aerr = fabsf(S0 - gt);
  // mean over HW of (p/pc - c/dg)^2, expanded
  const float sp = (S1 / (pc * pc) - 2.0f * S2 / (pc * dg) + S3 / (dg * dg))
                   * (1.0f / (float)HW);

  __shared__ float sh_a[BATCH], sh_s[BATCH];
  sh_a[t] = aerr; sh_s[t] = sp;
  __syncthreads();
  if (t == 0) {
    float ca = 0.f, cs = 0.f;
    for (int i = 0; i < BATCH; ++i) { ca += sh_a[i]; cs += sh_s[i]; }
    const float count_loss   = ca * (1.0f / BATCH);
    const float scale_loss   = count_loss / dg;   // mean(aerr/dg) == mean(aerr)/dg
    const float spatial_loss = cs * (1.0f / BATCH);
    out[0] = COUNT_W * count_loss + SPATIAL_W * spatial_loss + SCALE_W * scale_loss;
    out[1] = count_loss;
    out[2] = scale_loss;
    out[3] = spatial_loss;
  }
}

// ---------------- launcher ----------------
extern "C" void kernel_launch(void* const* d_in, const int* in_sizes, int n_in,
                              void* d_out, int out_size, void* d_ws, size_t ws_size,
                              hipStream_t stream) {
  const float* pred      = (const float*)d_in[0];
  const int*   points    = (const int*)d_in[1];
  const int*   downscale = (const int*)d_in[2];
  float*       out       = (float*)d_out;

  // workspace layout: [0, 32 MiB) packed uint16 histogram, then block partials
  unsigned* hist     = (unsigned*)d_ws;
  float*    partials = (float*)((char*)d_ws + (size_t)HIST_WORDS * sizeof(unsigned));

  // 1) zero histogram (harness poisons d_ws; re-zero every call)
  zero_hist_kernel<<<2048, 256, 0, stream>>>((uint4*)hist);

  // 2) scatter the 256k points
  const int npt = BATCH * NPTS;
  scatter_points_kernel<<<(npt + 255) / 256, 256, 0, stream>>>(points, downscale, hist);

  // 3) fused single pass over all pixels -> per-block partial sums
  fused_reduce_kernel<<<BATCH * BLOCKS_PER_BATCH, RED_THREADS, 0, stream>>>(
      pred, hist, partials);

  // 4) finalize the four output scalars
  finalize_kernel<<<1, BATCH, 0, stream>>>(partials, out);
}